// GCNLayer_38113539785115
// MI455X (gfx1250) — compile-verified
//
#include <hip/hip_runtime.h>
#include <hip/hip_bf16.h>

// GCN layer for MI455X (gfx1250, wave32, WMMA).
// y = BN(relu(segsum(h[src]->dst)*norm_dst + b) + relu(feature@res_W + res_b))
// with h = (feature@W)*norm_src  (row scaling commutes with the GEMM).

typedef __attribute__((ext_vector_type(2))) float v2f;
typedef __attribute__((ext_vector_type(8))) float v8f;

#define IN_F   128
#define OUT_F  64
#define BCOLS  128      // fused [W | res_W]
#define A_PITCH 132     // 128 + 4 padding floats -> conflict-free LDS reads

// ---------------------------------------------------------------- zero
__global__ void zero_kernel(float* __restrict__ p, long long n) {
    long long i = (long long)blockIdx.x * blockDim.x + threadIdx.x;
    if (i < n) p[i] = 0.0f;
}

// ---------------------------------------------------------------- degrees
__global__ void degree_kernel(const int* __restrict__ src, const int* __restrict__ dst,
                              float* __restrict__ out_deg, float* __restrict__ in_deg,
                              int nEdges) {
    int e = blockIdx.x * blockDim.x + threadIdx.x;
    if (e < nEdges) {
        atomicAdd(&out_deg[src[e]], 1.0f);
        atomicAdd(&in_deg[dst[e]], 1.0f);
    }
}

// deg -> rsqrt(max(deg,1)) in place (covers both norm arrays at once)
__global__ void norm_kernel(float* __restrict__ deg, int n) {
    int i = blockIdx.x * blockDim.x + threadIdx.x;
    if (i < n) deg[i] = rsqrtf(fmaxf(deg[i], 1.0f));
}

// ---------------------------------------------------------------- fused GEMM (WMMA f32 16x16x4)
// Block = 256 threads = 8 waves. Block computes 16 rows x 128 cols of
// Z = feature @ [W | res_W]. Wave w owns col tile [16w, 16w+16).
//   cols 0..63  -> h   = Z * norm_src[row]          (stored to ws)
//   cols 64..127-> res = relu(Z + res_b) -> d_out   (conv added later)
// Requires nNodes % 16 == 0 (true: 100000 = 6250*16).
__global__ void gemm_wmma_kernel(const float* __restrict__ feat,
                                 const float* __restrict__ W,
                                 const float* __restrict__ resW,
                                 const float* __restrict__ res_b,
                                 const float* __restrict__ norm_src,
                                 float* __restrict__ h,
                                 float* __restrict__ out) {
    __shared__ float As[16 * A_PITCH];

    const int rowBase = blockIdx.x * 16;
    const int t = threadIdx.x;

    // cooperative A-tile load: 16x128 floats, 8 consecutive floats per thread
    {
        const int r = t >> 4;             // 0..15
        const int c = (t & 15) * 8;       // 0,8,...,120
        const float4* g = (const float4*)(feat + (size_t)(rowBase + r) * IN_F + c);
        float4 v0 = g[0];
        float4 v1 = g[1];
        *(float4*)(As + r * A_PITCH + c)     = v0;
        *(float4*)(As + r * A_PITCH + c + 4) = v1;
    }
    __syncthreads();

    const int lane  = t & 31;
    const int wv    = t >> 5;             // wave id 0..7 -> col tile
    const int nloc  = lane & 15;          // matrix row (A) / col (B) within tile
    const int khalf = (lane >> 4) << 1;   // 0 for lanes 0-15, 2 for lanes 16-31
    const int col   = wv * 16 + nloc;     // fused column 0..127

    // B element (k, col): row-major [IN_F][OUT_F] for both W and res_W
    const float* Bcol = (col < OUT_F) ? (W + col) : (resW + (col - OUT_F));

    v8f acc = {};
    const float* Arow = As + nloc * A_PITCH;
#pragma unroll
    for (int k0 = 0; k0 < IN_F; k0 += 4) {
        const int k = k0 + khalf;
        // A fragment: 16x4 f32, lane holds (m=nloc, K=k),(m,K=k+1)
        v2f a = *(const v2f*)(Arow + k);              // 8B aligned (k even)
        // B fragment: 4x16 f32, lane holds (K=k,n=col),(K=k+1,n=col)
        v2f bfrag;
        bfrag.x = Bcol[(size_t)k * OUT_F];
        bfrag.y = Bcol[(size_t)(k + 1) * OUT_F];
        acc = __builtin_amdgcn_wmma_f32_16x16x4_f32(
                  /*neg_a=*/false, a, /*neg_b=*/false, bfrag,
                  /*c_mod=*/(short)0, acc, /*reuse_a=*/false, /*reuse_b=*/false);
    }

    // D layout: VGPR r holds row (r + 8*(lane>=16)), col = nloc of tile
    const int rbase = rowBase + ((lane >> 4) << 3);
#pragma unroll
    for (int r = 0; r < 8; ++r) {
        const int row = rbase + r;
        const float z = acc[r];
        if (col < OUT_F) {
            h[(size_t)row * OUT_F + col] = z * norm_src[row];
        } else {
            const int oc = col - OUT_F;
            out[(size_t)row * OUT_F + oc] = fmaxf(z + res_b[oc], 0.0f);
        }
    }
}

// ---------------------------------------------------------------- edge aggregation
// 64 threads per edge, one f32 atomic each; h-gather is coalesced.
__global__ void aggregate_kernel(const int* __restrict__ src, const int* __restrict__ dst,
                                 const float* __restrict__ h, float* __restrict__ agg,
                                 int nEdges) {
    long long tid = (long long)blockIdx.x * blockDim.x + threadIdx.x;
    int e = (int)(tid >> 6);
    int j = (int)(tid & 63);
    if (e < nEdges) {
        int s = src[e];
        int d = dst[e];
        atomicAdd(&agg[(size_t)d * OUT_F + j], h[(size_t)s * OUT_F + j]);
    }
}

// ---------------------------------------------------------------- combine + BN stats
// y = relu(agg*norm_dst + b) + res(out); accumulate per-feature sum / sumsq.
__global__ void combine_stats_kernel(const float* __restrict__ agg,
                                     const float* __restrict__ norm_dst,
                                     const float* __restrict__ bias,
                                     float* __restrict__ out,
                                     float* __restrict__ stats,  // [0..63]=sum, [64..127]=sumsq
                                     int nNodes) {
    __shared__ float s_sum[256];
    __shared__ float s_sq[256];
    const int j   = threadIdx.x & 63;
    const int off = threadIdx.x >> 6;     // 0..3
    const float bj = bias[j];
    float sum = 0.0f, sq = 0.0f;
    const int base = blockIdx.x * 64;
#pragma unroll
    for (int it = 0; it < 16; ++it) {
        const int node = base + it * 4 + off;
        if (node < nNodes) {
            const size_t idx = (size_t)node * OUT_F + j;
            const float conv = fmaxf(agg[idx] * norm_dst[node] + bj, 0.0f);
            const float y = conv + out[idx];
            out[idx] = y;
            sum += y;
            sq  += y * y;
        }
    }
    s_sum[threadIdx.x] = sum;
    s_sq[threadIdx.x]  = sq;
    __syncthreads();
    if (off == 0) {
        float ts = s_sum[j] + s_sum[j + 64] + s_sum[j + 128] + s_sum[j + 192];
        float tq = s_sq[j]  + s_sq[j + 64]  + s_sq[j + 128]  + s_sq[j + 192];
        atomicAdd(&stats[j], ts);
        atomicAdd(&stats[OUT_F + j], tq);
    }
}

// ---------------------------------------------------------------- BN normalize (in place)
__global__ void bn_kernel(float* __restrict__ out, const float* __restrict__ stats,
                          const float* __restrict__ gamma, const float* __restrict__ beta,
                          long long total, float invN) {
    long long idx = (long long)blockIdx.x * blockDim.x + threadIdx.x;
    if (idx < total) {
        const int j = (int)(idx & 63);
        const float mean = stats[j] * invN;
        const float var  = stats[OUT_F + j] * invN - mean * mean;
        out[idx] = (out[idx] - mean) * rsqrtf(var + 1e-5f) * gamma[j] + beta[j];
    }
}

// ---------------------------------------------------------------- launch
extern "C" void kernel_launch(void* const* d_in, const int* in_sizes, int n_in,
                              void* d_out, int out_size, void* d_ws, size_t ws_size,
                              hipStream_t stream) {
    const float* feature = (const float*)d_in[0];
    const int*   src     = (const int*)  d_in[1];
    const int*   dst     = (const int*)  d_in[2];
    const float* W       = (const float*)d_in[3];
    const float* b       = (const float*)d_in[4];
    const float* resW    = (const float*)d_in[5];
    const float* res_b   = (const float*)d_in[6];
    const float* gamma   = (const float*)d_in[7];
    const float* beta    = (const float*)d_in[8];
    float*       out     = (float*)d_out;

    const int nNodes = in_sizes[0] / IN_F;   // 100000
    const int nEdges = in_sizes[1];          // 1600000

    // workspace layout (floats): [norm_src N][norm_dst N][agg 64N][stats 128][h 64N]
    float* ws       = (float*)d_ws;
    float* norm_src = ws;
    float* norm_dst = ws + nNodes;
    float* agg      = ws + 2 * (size_t)nNodes;
    float* stats    = agg + (size_t)nNodes * OUT_F;
    float* h        = stats + 2 * OUT_F;

    // 1) zero degrees + agg + stats (contiguous prefix of ws)
    const long long zcount = 2LL * nNodes + (long long)nNodes * OUT_F + 2 * OUT_F;
    zero_kernel<<<(int)((zcount + 255) / 256), 256, 0, stream>>>(ws, zcount);

    // 2) degree accumulation
    degree_kernel<<<(nEdges + 255) / 256, 256, 0, stream>>>(src, dst, norm_src, norm_dst, nEdges);

    // 3) deg -> rsqrt(max(deg,1)) in place (both arrays)
    norm_kernel<<<(2 * nNodes + 255) / 256, 256, 0, stream>>>(norm_src, 2 * nNodes);

    // 4) fused WMMA GEMM: h (scaled) + residual relu into d_out
    gemm_wmma_kernel<<<nNodes / 16, 256, 0, stream>>>(feature, W, resW, res_b,
                                                      norm_src, h, out);

    // 5) edge scatter-add
    const long long aggThreads = (long long)nEdges * OUT_F;
    aggregate_kernel<<<(int)((aggThreads + 255) / 256), 256, 0, stream>>>(src, dst, h, agg, nEdges);

    // 6) combine conv + residual, accumulate BN stats
    combine_stats_kernel<<<(nNodes + 63) / 64, 256, 0, stream>>>(agg, norm_dst, b, out,
                                                                 stats, nNodes);

    // 7) batch-norm normalize in place
    const long long total = (long long)nNodes * OUT_F;
    bn_kernel<<<(int)((total + 255) / 256), 256, 0, stream>>>(out, stats, gamma, beta,
                                                              total, 1.0f / (float)nNodes);
}